// UnivariateTripleMLP_32160715112709
// MI455X (gfx1250) — compile-verified
//
#include <hip/hip_runtime.h>
#include <hip/hip_bf16.h>

typedef float v2f __attribute__((ext_vector_type(2)));
typedef float v8f __attribute__((ext_vector_type(8)));

#define NB   131072   // batch (columns of inputs)
#define K1   1024     // race cycle / layer-1 reduction dim
#define D1   64
#define D2   16
#define WAVES 4
#define TPB  (WAVES * 32)

// Layer 1 via V_WMMA_F32_16X16X4_F32 (exact fp32):
//   D[16b x 16d] += A(x tile 16b x 4k) * B(W1 tile 4k x 16d)
// A layout (ISA 7.12.2, 32-bit A 16x4): lane = M (b); VGPR0 = K {0|2}, VGPR1 = K {1|3}
// B layout (mirrored KxN): VGPR0 = row K {0|2} striped over lanes (N), VGPR1 = K {1|3}
// C/D layout: VGPR r -> rows M = r (lanes 0-15) and M = r+8 (lanes 16-31), N = lane%16
__global__ __launch_bounds__(TPB) void mlp3_wmma_f32_kernel(
    const float* __restrict__ inputs,  // [K1, NB] row-major
    const float* __restrict__ W1,      // [K1, D1]
    const float* __restrict__ b1,      // [D1]
    const float* __restrict__ W2,      // [D1, D2]
    const float* __restrict__ b2,      // [D2]
    const float* __restrict__ W3,      // [D2, 1]
    const float* __restrict__ b3,      // [1]
    float* __restrict__ out)           // [NB]
{
    __shared__ float h1s[WAVES][32][D1 + 1];   // +1 pad: conflict-free transpose

    const int tid  = threadIdx.x;
    const int wid  = tid >> 5;
    const int lane = tid & 31;
    const int half = lane >> 4;   // 0: K+0/1  1: K+2/3
    const int l16  = lane & 15;

    const int b_wave = blockIdx.x * (WAVES * 32) + wid * 32;  // 32 batch rows / wave

    // Per-lane base pointers for fragment loads
    const float* xcol = inputs + (size_t)(2 * half) * NB + (b_wave + l16); // A: x[b, k]
    const float* wrow = W1     + (size_t)(2 * half) * D1 + l16;           // B: W1[k, d]

    v8f acc[2][4];
    #pragma unroll
    for (int bt = 0; bt < 2; ++bt)
        #pragma unroll
        for (int dt = 0; dt < 4; ++dt) { v8f z = {}; acc[bt][dt] = z; }

    #pragma unroll 2
    for (int kb = 0; kb < K1; kb += 4) {
        const float* xr = xcol + (size_t)kb * NB;
        const float* wr = wrow + (size_t)kb * D1;

        // Stream-ahead prefetch of inputs (gfx1250 global_prefetch_b8)
        const int pk = (kb + 64 < K1) ? (kb + 64) : kb;
        __builtin_prefetch(xcol + (size_t)pk * NB, 0, 0);

        v2f a0, a1;                       // A fragments for the two b-tiles
        a0.x = xr[0];        a0.y = xr[NB];
        a1.x = xr[16];       a1.y = xr[NB + 16];

        v2f bf[4];                        // B fragments for the four d-tiles
        #pragma unroll
        for (int dt = 0; dt < 4; ++dt) {
            bf[dt].x = wr[dt * 16];
            bf[dt].y = wr[D1 + dt * 16];
        }

        #pragma unroll
        for (int dt = 0; dt < 4; ++dt) {
            acc[0][dt] = __builtin_amdgcn_wmma_f32_16x16x4_f32(
                false, a0, false, bf[dt], (short)0, acc[0][dt], false, false);
            acc[1][dt] = __builtin_amdgcn_wmma_f32_16x16x4_f32(
                false, a1, false, bf[dt], (short)0, acc[1][dt], false, false);
        }
    }

    // bias + ReLU, scatter h1 fragments into LDS (transpose staging)
    #pragma unroll
    for (int dt = 0; dt < 4; ++dt) {
        const float bias = b1[dt * 16 + l16];
        #pragma unroll
        for (int bt = 0; bt < 2; ++bt) {
            #pragma unroll
            for (int r = 0; r < 8; ++r) {
                float v = acc[bt][dt][r] + bias;
                v = fmaxf(v, 0.0f);
                h1s[wid][bt * 16 + r + 8 * half][dt * 16 + l16] = v;
            }
        }
    }
    __syncthreads();

    // Layers 2 + 3: one batch row per lane (tiny vs layer 1)
    float h2[D2];
    #pragma unroll
    for (int j = 0; j < D2; ++j) h2[j] = b2[j];

    const float* hrow = &h1s[wid][lane][0];
    #pragma unroll 4
    for (int k = 0; k < D1; ++k) {
        const float hk = hrow[k];
        #pragma unroll
        for (int j = 0; j < D2; ++j)
            h2[j] = fmaf(hk, W2[k * D2 + j], h2[j]);
    }

    float o = b3[0];
    #pragma unroll
    for (int j = 0; j < D2; ++j)
        o = fmaf(fmaxf(h2[j], 0.0f), W3[j], o);

    out[b_wave + lane] = o;
}

extern "C" void kernel_launch(void* const* d_in, const int* in_sizes, int n_in,
                              void* d_out, int out_size, void* d_ws, size_t ws_size,
                              hipStream_t stream) {
    const float* inputs = (const float*)d_in[0];
    const float* W1     = (const float*)d_in[1];
    const float* b1     = (const float*)d_in[2];
    const float* W2     = (const float*)d_in[3];
    const float* b2     = (const float*)d_in[4];
    const float* W3     = (const float*)d_in[5];
    const float* b3     = (const float*)d_in[6];
    float* out = (float*)d_out;

    const int grid = NB / (WAVES * 32);   // 1024 blocks x 128 threads
    mlp3_wmma_f32_kernel<<<grid, TPB, 0, stream>>>(inputs, W1, b1, W2, b2, W3, b3, out);
}